// AttnEncoder_55748675502366
// MI455X (gfx1250) — compile-verified
//
#include <hip/hip_runtime.h>
#include <math.h>

typedef __attribute__((ext_vector_type(2))) float v2f;
typedef __attribute__((ext_vector_type(8))) float v8f;

#define DFEAT 128
#define DDEG 32
#define NEG_SLOPE 0.2f

// ---------------------------------------------------------------------------
// Ordered-uint encoding of float for atomicMax-based segment max
// (monotone for all finite values incl. negatives).
// ---------------------------------------------------------------------------
__device__ __forceinline__ unsigned fenc(float f) {
  unsigned u = __float_as_uint(f);
  return (u & 0x80000000u) ? ~u : (u | 0x80000000u);
}
__device__ __forceinline__ float fdec(unsigned u) {
  u = (u & 0x80000000u) ? (u & 0x7fffffffu) : ~u;
  return __uint_as_float(u);
}

// ---------------------------------------------------------------------------
// h = X @ W   (X:[n,128] row-major, W:[128,128] row-major, h:[n,128])
// One block = 16 rows of X; 8 waves, each wave owns a 16-col tile of the
// output and runs 32 x V_WMMA_F32_16X16X4_F32 over K=128.
// A tile (16x128) staged in LDS with coalesced 32B-per-thread loads.
// ---------------------------------------------------------------------------
__global__ void __launch_bounds__(256) gat_gemm_xw(const float* __restrict__ X,
                                                   const float* __restrict__ W,
                                                   float* __restrict__ H,
                                                   int nrows) {
  __shared__ float xs[16 * DFEAT];  // 8 KB
  const int tid = threadIdx.x;
  const int wave = tid >> 5;  // 0..7 -> N tile
  const int lane = tid & 31;
  const int rowBlk = blockIdx.x * 16;

  // cooperative stage: 2048 floats, 8 per thread (two float4)
  {
    int idx = tid * 8;
    int r = idx >> 7;        // /128
    int c = idx & (DFEAT - 1);
    const float4* src = (const float4*)(X + (size_t)(rowBlk + r) * DFEAT + c);
    float4 v0 = src[0];
    float4 v1 = src[1];
    *(float4*)(xs + r * DFEAT + c) = v0;
    *(float4*)(xs + r * DFEAT + c + 4) = v1;
  }
  __syncthreads();

  const int col0 = wave * 16;
  const int mn = lane & 15;             // A: row M, B: col N
  const int kk = (lane >> 4) * 2;       // K sub-pair selected by lane half

  v8f acc = {};
  for (int k0 = 0; k0 < DFEAT; k0 += 4) {
    v2f a, b;
    a.x = xs[mn * DFEAT + k0 + kk];
    a.y = xs[mn * DFEAT + k0 + kk + 1];
    b.x = W[(size_t)(k0 + kk) * DFEAT + col0 + mn];
    b.y = W[(size_t)(k0 + kk + 1) * DFEAT + col0 + mn];
    acc = __builtin_amdgcn_wmma_f32_16x16x4_f32(
        /*neg_a=*/false, a, /*neg_b=*/false, b,
        /*c_mod=*/(short)0, acc, /*reuse_a=*/false, /*reuse_b=*/false);
  }

  const int mbase = (lane >> 4) * 8;
  float* out = H + (size_t)(rowBlk + mbase) * DFEAT + col0 + mn;
#pragma unroll
  for (int r = 0; r < 8; ++r) {
    out[(size_t)r * DFEAT] = acc[r];
  }
}

// ---------------------------------------------------------------------------
// Per-node attention pre-products: asrc[i] = h[i].a_src, adst[i] = h[i].a_dst
// One wave per node, 4 floats per lane, shfl tree reduction (wave32).
// ---------------------------------------------------------------------------
__global__ void __launch_bounds__(256) gat_node_alpha(const float* __restrict__ H,
                                                      const float* __restrict__ a_src,
                                                      const float* __restrict__ a_dst,
                                                      float* __restrict__ asrc,
                                                      float* __restrict__ adst,
                                                      int n) {
  int g = blockIdx.x * blockDim.x + threadIdx.x;
  int node = g >> 5;
  int lane = threadIdx.x & 31;
  if (node >= n) return;
  float4 h4 = *(const float4*)(H + (size_t)node * DFEAT + lane * 4);
  float4 s4 = *(const float4*)(a_src + lane * 4);
  float4 d4 = *(const float4*)(a_dst + lane * 4);
  float ps = h4.x * s4.x + h4.y * s4.y + h4.z * s4.z + h4.w * s4.w;
  float pd = h4.x * d4.x + h4.y * d4.y + h4.z * d4.z + h4.w * d4.w;
#pragma unroll
  for (int off = 16; off; off >>= 1) {
    ps += __shfl_down(ps, off, 32);
    pd += __shfl_down(pd, off, 32);
  }
  if (lane == 0) {
    asrc[node] = ps;
    adst[node] = pd;
  }
}

// init mmax to enc(-inf), denom to 0
__global__ void gat_init_seg(unsigned* __restrict__ mmax, float* __restrict__ denom, int n) {
  int i = blockIdx.x * blockDim.x + threadIdx.x;
  if (i >= n) return;
  mmax[i] = fenc(-INFINITY);
  denom[i] = 0.0f;
}

__global__ void fill_zero(float* __restrict__ p, int count) {
  int i = blockIdx.x * blockDim.x + threadIdx.x;
  if (i < count) p[i] = 0.0f;
}

// ---------------------------------------------------------------------------
// Edge pass 1: raw score + segment max (ordered-uint atomicMax).
// Edges [0,E) from index arrays, [E, E+n) are self loops.
// ---------------------------------------------------------------------------
__global__ void __launch_bounds__(256) gat_edge_score(const int* __restrict__ srcI,
                                                      const int* __restrict__ dstI,
                                                      int ecount, int n,
                                                      const float* __restrict__ asrc,
                                                      const float* __restrict__ adst,
                                                      float* __restrict__ esc,
                                                      unsigned* __restrict__ mmax) {
  int e = blockIdx.x * blockDim.x + threadIdx.x;
  if (e >= ecount + n) return;
  int s, d;
  if (e < ecount) { s = srcI[e]; d = dstI[e]; } else { s = d = e - ecount; }
  float v = asrc[s] + adst[d];
  v = v > 0.0f ? v : v * NEG_SLOPE;
  esc[e] = v;
  atomicMax(mmax + d, fenc(v));
}

// Edge pass 2: exp(score - max) + segment sum
__global__ void __launch_bounds__(256) gat_edge_exp(const int* __restrict__ dstI,
                                                    int ecount, int n,
                                                    float* __restrict__ esc,
                                                    const unsigned* __restrict__ mmax,
                                                    float* __restrict__ denom) {
  int e = blockIdx.x * blockDim.x + threadIdx.x;
  if (e >= ecount + n) return;
  int d = (e < ecount) ? dstI[e] : (e - ecount);
  float w = expf(esc[e] - fdec(mmax[d]));
  esc[e] = w;
  atomicAdd(denom + d, w);
}

// Edge pass 3: out[dst] += (w/denom) * h[src]; one wave per edge, 4 f32/lane.
__global__ void __launch_bounds__(256) gat_edge_scatter(const int* __restrict__ srcI,
                                                        const int* __restrict__ dstI,
                                                        int ecount, int n,
                                                        const float* __restrict__ esc,
                                                        const float* __restrict__ denom,
                                                        const float* __restrict__ H,
                                                        float* __restrict__ O) {
  int g = blockIdx.x * blockDim.x + threadIdx.x;
  int e = g >> 5;
  int lane = threadIdx.x & 31;
  if (e >= ecount + n) return;
  int s, d;
  if (e < ecount) { s = srcI[e]; d = dstI[e]; } else { s = d = e - ecount; }
  float coef = esc[e] / (denom[d] + 1e-16f);
  float4 h4 = *(const float4*)(H + (size_t)s * DFEAT + lane * 4);
  float* o = O + (size_t)d * DFEAT + lane * 4;
  atomicAdd(o + 0, coef * h4.x);
  atomicAdd(o + 1, coef * h4.y);
  atomicAdd(o + 2, coef * h4.z);
  atomicAdd(o + 3, coef * h4.w);
}

// epilogue: O += bias, optional relu
__global__ void __launch_bounds__(256) gat_bias_act(float* __restrict__ O,
                                                    const float* __restrict__ b,
                                                    int total, int relu) {
  int i = blockIdx.x * blockDim.x + threadIdx.x;
  if (i >= total) return;
  float v = O[i] + b[i & (DFEAT - 1)];
  if (relu) v = fmaxf(v, 0.0f);
  O[i] = v;
}

// ---------------------------------------------------------------------------
// Fusion head: logits = [x2o|x2s|deg] @ Wp + bp, 2-way softmax, blend.
// ---------------------------------------------------------------------------
__global__ void __launch_bounds__(256) gat_predict(const float* __restrict__ x2o,
                                                   const float* __restrict__ x2s,
                                                   const float* __restrict__ deg,
                                                   const float* __restrict__ Wp,
                                                   const float* __restrict__ bp,
                                                   float* __restrict__ out, int n) {
  int i = blockIdx.x * blockDim.x + threadIdx.x;
  if (i >= n) return;
  const float* zo = x2o + (size_t)i * DFEAT;
  const float* zs = x2s + (size_t)i * DFEAT;
  const float* zd = deg + (size_t)i * DDEG;
  float l0 = bp[0], l1 = bp[1];
#pragma unroll 4
  for (int d = 0; d < DFEAT; ++d) {
    l0 += zo[d] * Wp[d * 2 + 0];
    l1 += zo[d] * Wp[d * 2 + 1];
  }
#pragma unroll 4
  for (int d = 0; d < DFEAT; ++d) {
    l0 += zs[d] * Wp[(DFEAT + d) * 2 + 0];
    l1 += zs[d] * Wp[(DFEAT + d) * 2 + 1];
  }
#pragma unroll 4
  for (int d = 0; d < DDEG; ++d) {
    l0 += zd[d] * Wp[(2 * DFEAT + d) * 2 + 0];
    l1 += zd[d] * Wp[(2 * DFEAT + d) * 2 + 1];
  }
  float mx = fmaxf(l0, l1);
  float e0 = expf(l0 - mx), e1 = expf(l1 - mx);
  float inv = 1.0f / (e0 + e1);
  float a0 = e0 * inv, a1 = e1 * inv;
  float* po = out + (size_t)i * DFEAT;
#pragma unroll 4
  for (int d = 0; d < DFEAT; ++d) po[d] = a0 * zo[d] + a1 * zs[d];
}

// ---------------------------------------------------------------------------
// Host orchestration
// ---------------------------------------------------------------------------
static inline int ceil_div(long long a, long long b) { return (int)((a + b - 1) / b); }

static void run_gat_layer(const float* X, const int* srcI, const int* dstI, int ecount,
                          const float* W, const float* avs, const float* avd,
                          const float* bias, float* H, float* asrc, float* adst,
                          unsigned* mmax, float* denom, float* esc, float* O,
                          int n, int relu, hipStream_t stream) {
  // 1. h = X @ W  (WMMA)
  gat_gemm_xw<<<n / 16, 256, 0, stream>>>(X, W, H, n);
  // 2. per-node alpha dot products
  gat_node_alpha<<<ceil_div((long long)n * 32, 256), 256, 0, stream>>>(H, avs, avd, asrc, adst, n);
  // 3. init segment state + zero aggregation target
  gat_init_seg<<<ceil_div(n, 256), 256, 0, stream>>>(mmax, denom, n);
  fill_zero<<<ceil_div((long long)n * DFEAT, 256), 256, 0, stream>>>(O, n * DFEAT);
  // 4-6. edge passes
  int etot = ecount + n;
  gat_edge_score<<<ceil_div(etot, 256), 256, 0, stream>>>(srcI, dstI, ecount, n, asrc, adst, esc, mmax);
  gat_edge_exp<<<ceil_div(etot, 256), 256, 0, stream>>>(dstI, ecount, n, esc, mmax, denom);
  gat_edge_scatter<<<ceil_div((long long)etot * 32, 256), 256, 0, stream>>>(srcI, dstI, ecount, n,
                                                                            esc, denom, H, O);
  // 7. bias + activation
  gat_bias_act<<<ceil_div((long long)n * DFEAT, 256), 256, 0, stream>>>(O, bias, n * DFEAT, relu);
}

extern "C" void kernel_launch(void* const* d_in, const int* in_sizes, int n_in,
                              void* d_out, int out_size, void* d_ws, size_t ws_size,
                              hipStream_t stream) {
  // inputs in setup_inputs() dict order
  const float* x_o      = (const float*)d_in[0];
  const float* degree_o = (const float*)d_in[1];
  const int*   ei_o     = (const int*)d_in[2];   // [2,E] row-major
  const int*   ei_s     = (const int*)d_in[3];
  const float* W_o1 = (const float*)d_in[4];
  const float* as_o1 = (const float*)d_in[5];
  const float* ad_o1 = (const float*)d_in[6];
  const float* b_o1  = (const float*)d_in[7];
  const float* W_o2 = (const float*)d_in[8];
  const float* as_o2 = (const float*)d_in[9];
  const float* ad_o2 = (const float*)d_in[10];
  const float* b_o2  = (const float*)d_in[11];
  const float* W_s1 = (const float*)d_in[12];
  const float* as_s1 = (const float*)d_in[13];
  const float* ad_s1 = (const float*)d_in[14];
  const float* b_s1  = (const float*)d_in[15];
  const float* W_s2 = (const float*)d_in[16];
  const float* as_s2 = (const float*)d_in[17];
  const float* ad_s2 = (const float*)d_in[18];
  const float* b_s2  = (const float*)d_in[19];
  const float* W_pred = (const float*)d_in[20];
  const float* b_pred = (const float*)d_in[21];

  const int n = in_sizes[0] / DFEAT;      // 50000
  const int ecount = in_sizes[2] / 2;     // 1000000

  const int* src_o = ei_o;
  const int* dst_o = ei_o + ecount;
  const int* src_s = ei_s;
  const int* dst_s = ei_s + ecount;

  // workspace carve-up (floats)
  float* ws = (float*)d_ws;
  float* H     = ws;                       // n*128  gemm output
  float* X1    = H + (size_t)n * DFEAT;    // n*128  layer-1 output (reused)
  float* X2o   = X1 + (size_t)n * DFEAT;   // n*128
  float* X2s   = X2o + (size_t)n * DFEAT;  // n*128
  float* asrc  = X2s + (size_t)n * DFEAT;  // n
  float* adst  = asrc + n;                 // n
  unsigned* mmax = (unsigned*)(adst + n);  // n
  float* denom = (float*)(mmax + n);       // n
  float* esc   = denom + n;                // ecount + n

  // branch o
  run_gat_layer(x_o, src_o, dst_o, ecount, W_o1, as_o1, ad_o1, b_o1,
                H, asrc, adst, mmax, denom, esc, X1, n, /*relu=*/1, stream);
  run_gat_layer(X1, src_o, dst_o, ecount, W_o2, as_o2, ad_o2, b_o2,
                H, asrc, adst, mmax, denom, esc, X2o, n, /*relu=*/0, stream);
  // branch s
  run_gat_layer(x_o, src_s, dst_s, ecount, W_s1, as_s1, ad_s1, b_s1,
                H, asrc, adst, mmax, denom, esc, X1, n, /*relu=*/1, stream);
  run_gat_layer(X1, src_s, dst_s, ecount, W_s2, as_s2, ad_s2, b_s2,
                H, asrc, adst, mmax, denom, esc, X2s, n, /*relu=*/0, stream);

  // fusion head
  gat_predict<<<ceil_div(n, 256), 256, 0, stream>>>(X2o, X2s, degree_o, W_pred, b_pred,
                                                    (float*)d_out, n);
}